// LogPolarTransformV2_54485955117710
// MI455X (gfx1250) — compile-verified
//
#include <hip/hip_runtime.h>
#include <hip/hip_bf16.h>

// ---------------------------------------------------------------------------
// LogPolarTransformV2: 31x31 SAME conv of 12 single-channel 64x64 maps with a
// 128-filter bank, as implicit GEMM on the CDNA5 fp32 WMMA path.
//   A (filters) : M=128 x K=961   -> staged to LDS by the Tensor Data Mover
//   B (im2col)  : K=961 x N=4096  -> per-lane gather staged via VGPRs
// fp32 in / fp32 accumulate via V_WMMA_F32_16X16X4_F32 (matches fp32 ref).
// ---------------------------------------------------------------------------

typedef float v2f  __attribute__((ext_vector_type(2)));
typedef float v8f  __attribute__((ext_vector_type(8)));
typedef unsigned int u32x4 __attribute__((ext_vector_type(4)));
typedef int   i32x4 __attribute__((ext_vector_type(4)));
typedef int   i32x8 __attribute__((ext_vector_type(8)));

#define NTAU   16
#define NANG   8
#define NOC    128            // NTAU * NANG
#define HW_    64
#define KW     31
#define KTAPS  961            // 31*31
#define PADW   15

#define BM     64             // oc tile per workgroup
#define BN     64             // pixel tile per workgroup (= one output row)
#define BK     32             // K chunk staged in LDS
#define LSTR   34             // LDS row stride (floats) = BK + 2 pad DWORDs:
                              // even (8B-aligned f2 loads), 34*m mod 64 spreads
                              // the 16 half-wave lanes over distinct banks.
                              // Matches TDM pad_interval=4 (32 DW), pad_amount=1 (2 DW).

__global__ __launch_bounds__(256)
void logpolar_conv_wmma_f32(const float* __restrict__ inp,
                            const float* __restrict__ filt,
                            float* __restrict__ out)
{
    __shared__ float Asm[BM * LSTR];   // Asm[m][k]  filled by TENSOR_LOAD_TO_LDS
    __shared__ float Bsm[BN * LSTR];   // Bsm[x][k]  im2col gather

    const int tid  = threadIdx.x;
    const int lane = tid & 31;
    const int wave = tid >> 5;

    // grid: 12 images * 64 rows * 2 M-tiles = 1536 workgroups
    const int wg    = blockIdx.x;
    const int img   = wg >> 7;          // / 128
    const int rem   = wg & 127;
    const int yRow  = rem >> 1;         // output row handled by this WG
    const int mBase = (rem & 1) * BM;   // oc base (0 or 64)

    // 8 waves cover the 4x4 grid of 16x16 C tiles: wave -> (wm, wn0..wn0+1)
    const int wm  = wave & 3;
    const int wn0 = (wave >> 2) * 2;

    const float* inImg = inp + (size_t)img * (HW_ * HW_);

    v8f c0 = {};
    v8f c1 = {};

    // WMMA fragment lane mapping (ISA 7.12.2, 32-bit 16x4 A / 4x16 B):
    //   lanes 0-15 : K pair {0,1}, lanes 16-31 : K pair {2,3}; M/N = lane%16
    const int hi   = (lane >= 16) ? 1 : 0;
    const int koff = hi ? 2 : 0;
    const int l16  = lane & 15;
    const int mA   = wm * 16 + l16;          // A row this lane supplies
    const int nB0  = wn0 * 16 + l16;         // B column, tile 0
    const int nB1  = nB0 + 16;               // B column, tile 1

    // LDS byte offset of Asm (generic LDS addr truncates to LDS offset)
    const unsigned int ldsA = (unsigned int)(uintptr_t)(&Asm[0]);

    for (int kc = 0; kc < KTAPS; kc += BK) {
        // ---- TDM: stage filter tile  Asm[m][k] = filt[mBase+m][kc+k] ----
        // 2-D tile: tile_dim0 = BK (along k, stride 1), tile_dim1 = BM (rows,
        // stride 961). tensor_dim0 = remaining taps so the hardware OOB check
        // zero-fills the K tail. LDS padding 2 DWORDs per 32 -> stride 34.
        if (wave == 0) {
            unsigned long long gaddr =
                (unsigned long long)(uintptr_t)(filt + (size_t)mBase * KTAPS + kc);
            unsigned int tdim0 = (unsigned int)(KTAPS - kc);   // remaining taps
            unsigned int tdim1 = (unsigned int)(NOC - mBase);  // remaining rows

            u32x4 g0;
            g0[0] = 1u;                                   // count=1, user desc
            g0[1] = ldsA;                                 // lds_addr (bytes)
            g0[2] = (unsigned int)gaddr;                  // global_addr[31:0]
            g0[3] = (unsigned int)(gaddr >> 32) | (2u << 30); // addr[56:32] | type=2

            i32x8 g1;
            g1[0] = (int)((2u << 16)      // data_size = 4B
                        | (1u << 20)      // pad_enable
                        | (4u << 22)      // pad_interval: 32 DWORDs
                        | (1u << 25));    // pad_amount: 2 DWORDs
            g1[1] = (int)((tdim0 & 0xFFFFu) << 16);               // tensor_dim0[15:0]
            g1[2] = (int)(((tdim0 >> 16) & 0xFFFFu)               // tensor_dim0[31:16]
                        | ((tdim1 & 0xFFFFu) << 16));             // tensor_dim1[15:0]
            g1[3] = (int)(((tdim1 >> 16) & 0xFFFFu)               // tensor_dim1[31:16]
                        | ((unsigned)BK << 16));                  // tile_dim0 = 32
            g1[4] = (int)((unsigned)BM);                          // tile_dim1 = 64, tile_dim2 = 0
            g1[5] = (int)((unsigned)KTAPS);                       // tensor_dim0_stride[31:0] = 961
            g1[6] = 0;                                            // stride hi / dim1_stride lo
            g1[7] = 0;

            i32x4 gz4 = {0, 0, 0, 0};             // 2-D tensor: groups 2/3 unused
            i32x8 gz8 = {0, 0, 0, 0, 0, 0, 0, 0}; // extra group in 6-arg builtin: unused
            __builtin_amdgcn_tensor_load_to_lds(g0, g1, gz4, gz4, gz8, 0);
        }

        // ---- stage im2col chunk:  Bsm[x][kk] = in[yRow+ky-15][x+kx-15] ----
        // (all 8 waves gather while wave 0's TDM transfer is in flight)
        #pragma unroll
        for (int j = 0; j < 8; ++j) {
            int e  = tid + 256 * j;          // 0..2047
            int kk = e >> 6;                 // 0..31
            int x  = e & 63;
            int k  = kc + kk;
            float v = 0.0f;
            if (k < KTAPS) {
                int ky = k / KW;
                int kx = k - ky * KW;
                int iy = yRow + ky - PADW;
                int ix = x + kx - PADW;
                if ((unsigned)iy < (unsigned)HW_ && (unsigned)ix < (unsigned)HW_)
                    v = inImg[iy * HW_ + ix];
            }
            Bsm[x * LSTR + kk] = v;
        }

        if (wave == 0)
            __builtin_amdgcn_s_wait_tensorcnt(0);   // A tile landed in LDS
        __syncthreads();

        // ---- 8 x (K=4) fp32 WMMA steps, 2 N-tiles sharing the A fragment ----
        #pragma unroll
        for (int kk = 0; kk < BK; kk += 4) {
            v2f a  = *(const v2f*)&Asm[mA  * LSTR + kk + koff];
            v2f b0 = *(const v2f*)&Bsm[nB0 * LSTR + kk + koff];
            v2f b1 = *(const v2f*)&Bsm[nB1 * LSTR + kk + koff];
            c0 = __builtin_amdgcn_wmma_f32_16x16x4_f32(
                     false, a, false, b0, (short)0, c0, false, false);
            c1 = __builtin_amdgcn_wmma_f32_16x16x4_f32(
                     false, a, false, b1, (short)0, c1, false, false);
        }
        __syncthreads();
    }

    // ---- store: C VGPR j -> M=j (+8 for hi half), N=lane%16.
    // Reference output layout (B, F, tau, theta, H, W) with oc = theta*16+tau.
    const int x0 = wn0 * 16 + l16;
    #pragma unroll
    for (int j = 0; j < 8; ++j) {
        int oc    = mBase + wm * 16 + (hi ? 8 : 0) + j;
        int tau   = oc & (NTAU - 1);
        int theta = oc >> 4;
        size_t base = ((((size_t)img * NTAU + tau) * NANG + theta) * (HW_ * HW_))
                      + (size_t)yRow * HW_;
        out[base + x0]      = c0[j];
        out[base + x0 + 16] = c1[j];
    }
}

extern "C" void kernel_launch(void* const* d_in, const int* in_sizes, int n_in,
                              void* d_out, int out_size, void* d_ws, size_t ws_size,
                              hipStream_t stream) {
    (void)in_sizes; (void)n_in; (void)d_ws; (void)ws_size; (void)out_size;
    const float* inp  = (const float*)d_in[0];   // (4,3,64,64)  fp32
    const float* filt = (const float*)d_in[1];   // (128,1,31,31) fp32
    float*       out  = (float*)d_out;           // (4,3,16,8,64,64) fp32

    const int numWG = 12 * 64 * 2;               // img * row * M-tiles = 1536
    logpolar_conv_wmma_f32<<<numWG, 256, 0, stream>>>(inp, filt, out);
}